// DCTL2D_14181982012168
// MI455X (gfx1250) — compile-verified
//
#include <hip/hip_runtime.h>

typedef float v2f __attribute__((ext_vector_type(2)));
typedef float v4f __attribute__((ext_vector_type(4)));
typedef float v8f __attribute__((ext_vector_type(8)));

#define WAVES_PER_BLOCK 8
#define LDS_STRIDE 20   // 16 + 4 pad floats: 20*N mod 64 distinct -> conflict-free

// out = C X C^T per 8x8 block. 4 blocks packed per 16x16 tile, sandwiched with
// Abd = diag(C,C) via V_WMMA_F32_16X16X4_F32 chains:
//   T = Abd @ X ; V = Abd @ T^T ; out = V^T (transpose folded into store addrs).
__global__ __launch_bounds__(256) void dct2d_wmma_kernel(
    const float* __restrict__ x, const float* __restrict__ Cm,
    float* __restrict__ out, int ntiles) {
  __shared__ float lds[WAVES_PER_BLOCK][16 * LDS_STRIDE];

  const int lane = threadIdx.x & 31;
  const int wave = threadIdx.x >> 5;
  float* buf = lds[wave];

  const int g = lane >> 4;   // half-wave group (A/B lane-group K offset = 2g)
  const int N = lane & 15;   // this lane's column (B/C/D) == row (A) index

  // ---- A fragments: Abd = blockdiag(C, C); 4 K-slices of 4.
  // ISA A 16x4 f32 layout: VGPR0 holds K=4s+2g, VGPR1 holds K=4s+2g+1, M = lane%16.
  v2f a[4];
#pragma unroll
  for (int s = 0; s < 4; ++s) {
    const int k0 = 4 * s + 2 * g;
    const int k1 = k0 + 1;
    const float c0 = Cm[(N & 7) * 8 + (k0 & 7)];
    const float c1 = Cm[(N & 7) * 8 + (k1 & 7)];
    a[s].x = ((k0 >> 3) == (N >> 3)) ? c0 : 0.0f;
    a[s].y = ((k1 >> 3) == (N >> 3)) ? c1 : 0.0f;
  }

  const int nwaves = (int)gridDim.x * WAVES_PER_BLOCK;
  // readfirstlane -> SGPR-uniform loop bounds, EXEC stays all-ones (WMMA req.)
  int tile = __builtin_amdgcn_readfirstlane((int)blockIdx.x * WAVES_PER_BLOCK + wave);

  for (; tile < ntiles; tile += nwaves) {
    const float* xt = x + (size_t)tile * 256;
    float* ot = out + (size_t)tile * 256;

    if (tile + nwaves < ntiles)  // uniform condition
      __builtin_prefetch(x + (size_t)(tile + nwaves) * 256 + lane * 8, 0, 0);

    // ---- load 4 blocks (1KB, coalesced b128) and scatter transposed to LDS:
    // buf[Ncol*STRIDE + Krow] = X_tile[Krow][Ncol]; quadrant (a,b) = block 2a+b.
    {
      const v4f lo = *(const v4f*)(xt + lane * 8);
      const v4f hi = *(const v4f*)(xt + lane * 8 + 4);
      const int q = lane >> 3;            // source block 0..3
      const int j = lane & 7;             // row within block
      const int K = 8 * (q >> 1) + j;     // tile row
      const int nb = 8 * (q & 1);         // tile col base
      const float t[8] = {lo.x, lo.y, lo.z, lo.w, hi.x, hi.y, hi.z, hi.w};
#pragma unroll
      for (int k = 0; k < 8; ++k)
        buf[(nb + k) * LDS_STRIDE + K] = t[k];
    }

    // ---- pass 1: T = Abd @ X.  B 4x16 layout: lane group g holds rows 4s+2g,+1
    v2f b[4];
#pragma unroll
    for (int s = 0; s < 4; ++s)
      b[s] = *(const v2f*)(buf + N * LDS_STRIDE + 4 * s + 2 * g);

    v8f acc = {};
#pragma unroll
    for (int s = 0; s < 4; ++s)
      acc = __builtin_amdgcn_wmma_f32_16x16x4_f32(false, a[s], false, b[s],
                                                  (short)0, acc, false, false);

    // C/D layout: VGPR r holds T[r+8g][N]. Store T row-major: buf[M*STRIDE+N].
#pragma unroll
    for (int r = 0; r < 8; ++r)
      buf[(r + 8 * g) * LDS_STRIDE + N] = acc[r];

    // ---- pass 2: V = Abd @ T^T.  (T^T)[K][N] = buf[N*STRIDE + K] -> same read pattern.
#pragma unroll
    for (int s = 0; s < 4; ++s)
      b[s] = *(const v2f*)(buf + N * LDS_STRIDE + 4 * s + 2 * g);

    v8f acc2 = {};
#pragma unroll
    for (int s = 0; s < 4; ++s)
      acc2 = __builtin_amdgcn_wmma_f32_16x16x4_f32(false, a[s], false, b[s],
                                                   (short)0, acc2, false, false);

    // out = V^T: lane's VGPR r = V[r+8g][N] = out[N][r+8g]
    //  -> block 2*(N/8)+g, element (N%8)*8 + r : 8 consecutive floats / lane.
    float* oblk = ot + (2 * (N >> 3) + g) * 64 + (N & 7) * 8;
    const v4f s0 = {acc2[0], acc2[1], acc2[2], acc2[3]};
    const v4f s1 = {acc2[4], acc2[5], acc2[6], acc2[7]};
    *(v4f*)(oblk) = s0;
    *(v4f*)(oblk + 4) = s1;
  }
}

extern "C" void kernel_launch(void* const* d_in, const int* in_sizes, int n_in,
                              void* d_out, int out_size, void* d_ws, size_t ws_size,
                              hipStream_t stream) {
  const float* x  = (const float*)d_in[0];
  const float* Cm = (const float*)d_in[1];
  float* out = (float*)d_out;

  const int nblk = in_sizes[0] / 64;   // number of 8x8 blocks (524288)
  const int ntiles = nblk / 4;         // 4 blocks per 16x16 tile (131072)

  int grid = 4096;                     // 32768 waves -> 4 tiles/wave
  const int maxgrid = (ntiles + WAVES_PER_BLOCK - 1) / WAVES_PER_BLOCK;
  if (grid > maxgrid) grid = maxgrid;
  if (grid < 1) grid = 1;

  dct2d_wmma_kernel<<<grid, 256, 0, stream>>>(x, Cm, out, ntiles);
}